// Fusion_44504451121883
// MI455X (gfx1250) — compile-verified
//
#include <hip/hip_runtime.h>
#include <math.h>

#define N_K 20000
#define N_E 50000
#define N_U 100000
#define EMB 128
#define E_UND 200000
#define E_EK 400000
#define E_EU 600000

typedef float v2f __attribute__((ext_vector_type(2)));
typedef float v8f __attribute__((ext_vector_type(8)));

// ---------------------------------------------------------------------------
// z[n,128] = concat(h1[0:split], h2[0:n-split]) @ W[128,128]  (fp32 WMMA)
// One block = 256 threads = 8 waves; wave w computes the 16x16 tile at
// rows [blockIdx.x*16, +16), cols [w*16, +16), K-loop of 32 x 16x16x4 WMMA.
// A frag (16x4 fp32): lanes 0-15 hold M=lane K={k0,k0+1}; lanes 16-31 K={k0+2,k0+3}.
// B frag (4x16 fp32): lanes 0-15 hold N=lane K={k0,k0+1}; lanes 16-31 K={k0+2,k0+3}.
// C/D (16x16 fp32): VGPR j, lane l -> (M = j + 8*(l>=16), N = l&15).
// ---------------------------------------------------------------------------
__global__ void gemm_zW(const float* __restrict__ h1, const float* __restrict__ h2,
                        int split, int n, const float* __restrict__ W,
                        float* __restrict__ z) {
  const int lane = threadIdx.x & 31;
  const int wave = threadIdx.x >> 5;      // column tile 0..7
  const int row0 = blockIdx.x * 16;
  const int n0 = wave * 16;
  const int m = lane & 15;
  const int khi = (lane >> 4) << 1;       // 0 or 2

  int arow = row0 + m;
  if (arow >= n) arow = n - 1;            // clamp loads; stores are guarded
  const float* ap = (arow < split) ? (h1 + (size_t)arow * EMB)
                                   : (h2 + (size_t)(arow - split) * EMB);
  const float* bp = W + n0 + m;

  v8f acc = {};
#pragma unroll
  for (int k0 = 0; k0 < EMB; k0 += 4) {
    v2f a, b;
    a[0] = ap[k0 + khi];
    a[1] = ap[k0 + khi + 1];
    b[0] = bp[(size_t)(k0 + khi) * EMB];
    b[1] = bp[(size_t)(k0 + khi + 1) * EMB];
    acc = __builtin_amdgcn_wmma_f32_16x16x4_f32(false, a, false, b,
                                                (short)0, acc, false, false);
  }
  const int rhalf = (lane >> 4) * 8;
#pragma unroll
  for (int j = 0; j < 8; ++j) {
    int r = row0 + rhalf + j;
    if (r < n) z[(size_t)r * EMB + n0 + m] = acc[j];
  }
}

// ---------------------------------------------------------------------------
// el[i] = z[i,:] . a[0:128] ; er[i] = z[i,:] . a[128:256]  (wave per row)
// ---------------------------------------------------------------------------
__global__ void row_dots(const float* __restrict__ z, const float* __restrict__ a,
                         float* __restrict__ el, float* __restrict__ er, int n) {
  const int lane = threadIdx.x & 31;
  const int row = blockIdx.x * (blockDim.x >> 5) + (threadIdx.x >> 5);
  if (row >= n) return;
  const float4 zv = *(const float4*)(z + (size_t)row * EMB + lane * 4);
  const float4 al = *(const float4*)(a + lane * 4);
  const float4 ar = *(const float4*)(a + EMB + lane * 4);
  float pl = zv.x * al.x + zv.y * al.y + zv.z * al.z + zv.w * al.w;
  float pr = zv.x * ar.x + zv.y * ar.y + zv.z * ar.z + zv.w * ar.w;
#pragma unroll
  for (int off = 16; off; off >>= 1) {
    pl += __shfl_xor(pl, off, 32);
    pr += __shfl_xor(pr, off, 32);
  }
  if (lane == 0) { el[row] = pl; er[row] = pr; }
}

// Order-preserving float <-> uint32 (for atomic segment-max on float).
__device__ __forceinline__ unsigned f2ord(float f) {
  int i = __float_as_int(f);
  return (i >= 0) ? ((unsigned)i | 0x80000000u) : ~(unsigned)i;
}
__device__ __forceinline__ float ord2f(unsigned u) {
  return (u & 0x80000000u) ? __int_as_float((int)(u & 0x7fffffffu))
                           : __uint_as_float(~u);
}

// e = leaky_relu(el[src]+er[dst], 0.01); segment max over dst (atomic u32 max)
__global__ void edge_max(const int* __restrict__ src, const int* __restrict__ dst,
                         int soff, int doff, int dbase,
                         const float* __restrict__ el, const float* __restrict__ er,
                         float* __restrict__ eval, unsigned* __restrict__ mu, int nE) {
  int e = blockIdx.x * blockDim.x + threadIdx.x;
  if (e >= nE) return;
  int s = src[e] + soff;
  int d = dst[e] + doff;
  float x = el[s] + er[d];
  x = (x > 0.f) ? x : 0.01f * x;
  eval[e] = x;
  atomicMax(mu + (d - dbase), f2ord(x));
}

// decode max (empty segment -> 0, matching where(isfinite)); zero the sum acc
__global__ void finalize_max(unsigned* __restrict__ mu, float* __restrict__ s, int n) {
  int i = blockIdx.x * blockDim.x + threadIdx.x;
  if (i >= n) return;
  unsigned u = mu[i];
  ((float*)mu)[i] = (u == 0u) ? 0.f : ord2f(u);
  s[i] = 0.f;
}

// ex = exp(e - m[dst]); segment sum over dst (native f32 atomic add)
__global__ void edge_exp(const int* __restrict__ dst, int doff, int dbase,
                         const float* __restrict__ eval, const float* __restrict__ mf,
                         float* __restrict__ exv, float* __restrict__ s, int nE) {
  int e = blockIdx.x * blockDim.x + threadIdx.x;
  if (e >= nE) return;
  int d = dst[e] + doff - dbase;
  float ex = __expf(eval[e] - mf[d]);
  exv[e] = ex;
  unsafeAtomicAdd(s + d, ex);
}

// out[dst,:] += (ex/s[dst]) * z[src,:]   (wave per edge, float4 per lane)
__global__ void edge_scatter(const int* __restrict__ src, const int* __restrict__ dst,
                             int soff, int doff, int dbase,
                             const float* __restrict__ exv, const float* __restrict__ s,
                             const float* __restrict__ z, float* __restrict__ out, int nE) {
  const int lane = threadIdx.x & 31;
  const int e = blockIdx.x * (blockDim.x >> 5) + (threadIdx.x >> 5);
  if (e >= nE) return;
  const int si = src[e] + soff;
  const int d = dst[e] + doff - dbase;
  const float alpha = exv[e] / s[d];
  const float4 zv = *(const float4*)(z + (size_t)si * EMB + lane * 4);
  float* o = out + (size_t)d * EMB + lane * 4;
  unsafeAtomicAdd(o + 0, alpha * zv.x);
  unsafeAtomicAdd(o + 1, alpha * zv.y);
  unsafeAtomicAdd(o + 2, alpha * zv.z);
  unsafeAtomicAdd(o + 3, alpha * zv.w);
}

// ---------------------------------------------------------------------------
// out = A + g0*X + g1*Y  where [g0,g1] = softmax([A.wx_lo + X.wx_hi + bx,
//                                                 A.wy_lo + Y.wy_hi + by])
// wave per row; xor-reduce so every lane holds the full dot products.
// ---------------------------------------------------------------------------
__global__ void fuse2(const float* __restrict__ A, const float* __restrict__ X,
                      const float* __restrict__ Y,
                      const float* __restrict__ wx, const float* __restrict__ bx,
                      const float* __restrict__ wy, const float* __restrict__ by,
                      float* __restrict__ out, int n) {
  const int lane = threadIdx.x & 31;
  const int row = blockIdx.x * (blockDim.x >> 5) + (threadIdx.x >> 5);
  if (row >= n) return;
  const size_t base = (size_t)row * EMB + lane * 4;
  const float4 a = *(const float4*)(A + base);
  const float4 x = *(const float4*)(X + base);
  const float4 y = *(const float4*)(Y + base);
  const float4 wxl = *(const float4*)(wx + lane * 4);
  const float4 wxh = *(const float4*)(wx + EMB + lane * 4);
  const float4 wyl = *(const float4*)(wy + lane * 4);
  const float4 wyh = *(const float4*)(wy + EMB + lane * 4);
  float px = a.x * wxl.x + a.y * wxl.y + a.z * wxl.z + a.w * wxl.w
           + x.x * wxh.x + x.y * wxh.y + x.z * wxh.z + x.w * wxh.w;
  float py = a.x * wyl.x + a.y * wyl.y + a.z * wyl.z + a.w * wyl.w
           + y.x * wyh.x + y.y * wyh.y + y.z * wyh.z + y.w * wyh.w;
#pragma unroll
  for (int off = 16; off; off >>= 1) {
    px += __shfl_xor(px, off, 32);
    py += __shfl_xor(py, off, 32);
  }
  const float sx = px + bx[0];
  const float sy = py + by[0];
  const float mx = fmaxf(sx, sy);
  const float exx = __expf(sx - mx);
  const float exy = __expf(sy - mx);
  const float inv = 1.f / (exx + exy);
  const float gx = exx * inv, gy = exy * inv;
  float4 o;
  o.x = a.x + gx * x.x + gy * y.x;
  o.y = a.y + gx * x.y + gy * y.y;
  o.z = a.z + gx * x.z + gy * y.z;
  o.w = a.w + gx * x.w + gy * y.w;
  *(float4*)(out + base) = o;
}

// ---------------------------------------------------------------------------
static void run_gat(hipStream_t stream,
                    const float* h1, const float* h2, int split, int n_nodes,
                    const float* W, const float* a,
                    const int* src, const int* dst, int soff, int doff,
                    int nE, int dbase, int n_dst,
                    float* z, float* el, float* er, unsigned* mu, float* s,
                    float* eval, float* exv,
                    float* out, bool zero_out) {
  if (zero_out)
    hipMemsetAsync(out, 0, (size_t)n_dst * EMB * sizeof(float), stream);
  hipMemsetAsync(mu, 0, (size_t)n_dst * sizeof(unsigned), stream);
  gemm_zW<<<(n_nodes + 15) / 16, 256, 0, stream>>>(h1, h2, split, n_nodes, W, z);
  row_dots<<<(n_nodes + 7) / 8, 256, 0, stream>>>(z, a, el, er, n_nodes);
  edge_max<<<(nE + 255) / 256, 256, 0, stream>>>(src, dst, soff, doff, dbase,
                                                 el, er, eval, mu, nE);
  finalize_max<<<(n_dst + 255) / 256, 256, 0, stream>>>(mu, s, n_dst);
  edge_exp<<<(nE + 255) / 256, 256, 0, stream>>>(dst, doff, dbase, eval,
                                                 (const float*)mu, exv, s, nE);
  edge_scatter<<<(nE + 7) / 8, 256, 0, stream>>>(src, dst, soff, doff, dbase,
                                                 exv, s, z, out, nE);
}

extern "C" void kernel_launch(void* const* d_in, const int* in_sizes, int n_in,
                              void* d_out, int out_size, void* d_ws, size_t ws_size,
                              hipStream_t stream) {
  (void)in_sizes; (void)n_in; (void)out_size; (void)ws_size;
  const float* kn_emb   = (const float*)d_in[0];
  const float* exer_emb = (const float*)d_in[1];
  const float* stu_emb  = (const float*)d_in[2];
  const int* und_src = (const int*)d_in[3];
  const int* und_dst = (const int*)d_in[4];
  const int* ek_src  = (const int*)d_in[5];
  const int* ek_dst  = (const int*)d_in[6];
  const int* eu_src  = (const int*)d_in[7];
  const int* eu_dst  = (const int*)d_in[8];
  const float* W_und = (const float*)d_in[9];
  const float* a_und = (const float*)d_in[10];
  const float* W_ek  = (const float*)d_in[11];
  const float* a_ek  = (const float*)d_in[12];
  const float* W_ke  = (const float*)d_in[13];
  const float* a_ke  = (const float*)d_in[14];
  const float* W_eu  = (const float*)d_in[15];
  const float* a_eu  = (const float*)d_in[16];
  const float* W_ue  = (const float*)d_in[17];
  const float* a_ue  = (const float*)d_in[18];
  const float* k_fc2_w = (const float*)d_in[19];
  const float* k_fc2_b = (const float*)d_in[20];
  const float* k_fc3_w = (const float*)d_in[21];
  const float* k_fc3_b = (const float*)d_in[22];
  const float* e_fc1_w = (const float*)d_in[23];
  const float* e_fc1_b = (const float*)d_in[24];
  const float* e_fc2_w = (const float*)d_in[25];
  const float* e_fc2_b = (const float*)d_in[26];

  float* outp = (float*)d_out;
  float* kn_out   = outp;
  float* exer_out = outp + (size_t)N_K * EMB;
  float* stu_out  = outp + (size_t)(N_K + N_E) * EMB;

  // workspace carve-up (~156 MB)
  float* p = (float*)d_ws;
  float* z  = p; p += (size_t)(N_E + N_U) * EMB;   // 150000*128
  float* C  = p; p += (size_t)N_K * EMB;           // k_und out
  float* D  = p; p += (size_t)N_K * EMB;           // e2k out (kn rows)
  float* B  = p; p += (size_t)N_E * EMB;           // k2e out (exer rows)
  float* Ce = p; p += (size_t)N_E * EMB;           // u2e out (exer rows)
  float* el = p; p += (size_t)(N_E + N_U);
  float* er = p; p += (size_t)(N_E + N_U);
  unsigned* mu = (unsigned*)p; p += (size_t)(N_E + N_U);
  float* s  = p; p += (size_t)(N_E + N_U);
  float* ev = p; p += (size_t)E_EU;
  float* ex = p; p += (size_t)E_EU;

  // 1) k_und : kn -> kn
  run_gat(stream, kn_emb, kn_emb, N_K, N_K, W_und, a_und,
          und_src, und_dst, 0, 0, E_UND, 0, N_K,
          z, el, er, mu, s, ev, ex, C, true);
  // 2) e2k : nodes = [exer | kn], dst = ek_dst + N_E
  run_gat(stream, exer_emb, kn_emb, N_E, N_E + N_K, W_ek, a_ek,
          ek_src, ek_dst, 0, N_E, E_EK, N_E, N_K,
          z, el, er, mu, s, ev, ex, D, true);
  // 3) k2e : src = ek_dst + N_E, dst = ek_src
  run_gat(stream, exer_emb, kn_emb, N_E, N_E + N_K, W_ke, a_ke,
          ek_dst, ek_src, N_E, 0, E_EK, 0, N_E,
          z, el, er, mu, s, ev, ex, B, true);
  // 4) e2u : nodes = [exer | stu]; accumulate directly into stu_out
  hipMemcpyAsync(stu_out, stu_emb, (size_t)N_U * EMB * sizeof(float),
                 hipMemcpyDeviceToDevice, stream);
  run_gat(stream, exer_emb, stu_emb, N_E, N_E + N_U, W_eu, a_eu,
          eu_src, eu_dst, 0, N_E, E_EU, N_E, N_U,
          z, el, er, mu, s, ev, ex, stu_out, false);
  // 5) u2e : src = eu_dst + N_E, dst = eu_src
  run_gat(stream, exer_emb, stu_emb, N_E, N_E + N_U, W_ue, a_ue,
          eu_dst, eu_src, N_E, 0, E_EU, 0, N_E,
          z, el, er, mu, s, ev, ex, Ce, true);

  // fusion heads
  fuse2<<<(N_K + 7) / 8, 256, 0, stream>>>(kn_emb, C, D,
                                           k_fc2_w, k_fc2_b, k_fc3_w, k_fc3_b,
                                           kn_out, N_K);
  fuse2<<<(N_E + 7) / 8, 256, 0, stream>>>(exer_emb, B, Ce,
                                           e_fc1_w, e_fc1_b, e_fc2_w, e_fc2_b,
                                           exer_out, N_E);
}